// CustomEmbedding_82514911691024
// MI455X (gfx1250) — compile-verified
//
#include <hip/hip_runtime.h>
#include <hip/hip_bf16.h>
#include <cstdint>
#include <cstddef>

// ---------------------------------------------------------------------------
// CustomEmbedding for MI455X (gfx1250):
//   out[tok][:] = (id < 1000) ? softsign_powers(id) : W[id][:]
//
// Roofline: 256MB output stream + ~251MB W gather. W (50257*512*4B = 103MB)
// fits in the 192MB L2, so with NT (non-temporal) output stores the steady
// state is: gather reads hit L2, HBM traffic ~= 256MB of writes
// -> ~11us floor at 23.3 TB/s. Compute (softsign powers) is negligible VALU.
//
// CDNA5 path: the 2KB W-row copy is done as an async global->LDS->global DMA
// (GLOBAL_LOAD_ASYNC_TO_LDS_B128 / GLOBAL_STORE_ASYNC_FROM_LDS_B128, tracked
// by ASYNCcnt) — row data never passes through VGPRs, so waves carry no
// load-return register dependencies. The async store uses th:TH_STORE_NT so
// the output stream does not evict W from L2.
// ---------------------------------------------------------------------------

#define EMB_DIM   512
#define NUM_CNT   1000
#define MEAN_F    499.5f
#define STD_F     288.6749912572095f   /* float(np.std(arange(1000)) + 1e-6) */

typedef float f4 __attribute__((ext_vector_type(4)));
typedef __attribute__((address_space(3))) float lds_float;

__global__ __launch_bounds__(256) void embed_kernel(const int* __restrict__ x,
                                                    const float* __restrict__ W,
                                                    float* __restrict__ out,
                                                    int n_tokens) {
    __shared__ __align__(16) float smem[2 * EMB_DIM];   // 4KB staging, 2 tokens/block

    const int tid  = threadIdx.x;
    const int half = tid >> 7;        // token slot within block (0/1)
    const int q    = tid & 127;       // float4 slot within the 512-dim row
    const int tok  = blockIdx.x * 2 + half;
    if (tok >= n_tokens) return;

    const int id = x[tok];            // same addr across 128 lanes -> broadcast
    float* dst = out + (size_t)tok * EMB_DIM + q * 4;

    if (id < NUM_CNT) {
        // analytic numeric embedding: s^(d+1), d = 4q .. 4q+3
        const float nrm = ((float)id - MEAN_F) / STD_F;
        const float s   = nrm / (1.0f + fabsf(nrm));     // softsign
        const float s2  = s * s;
        const float s4  = s2 * s2;
        float p = 1.0f, b = s4;
        int   e = q;                                     // q in [0,127], 7 bits
        #pragma unroll
        for (int i = 0; i < 7; ++i) {                    // p = (s^4)^q
            if (e & 1) p *= b;
            b *= b;
            e >>= 1;
        }
        f4 v;
        v.x = p * s;                                     // s^(4q+1)
        v.y = v.x * s;
        v.z = v.y * s;
        v.w = v.z * s;
        __builtin_nontemporal_store(v, (f4*)dst);        // NT: stream, keep W in L2
    } else {
        // learned embedding: DMA the 2KB row global -> LDS -> global.
        // Load: default RT temporal hint (W stays L2-resident).
        // Store: NT temporal hint (output streams past L2).
        const float* src = W + (size_t)id * EMB_DIM + q * 4;
        lds_float*   lp  = (lds_float*)&smem[half * EMB_DIM + q * 4];

        asm volatile("global_load_async_to_lds_b128 %0, %1, off"
                     :: "v"(lp), "v"(src) : "memory");
        asm volatile("s_wait_asynccnt 0x0" ::: "memory");   // LDS data landed
        asm volatile("global_store_async_from_lds_b128 %0, %1, off th:TH_STORE_NT"
                     :: "v"(dst), "v"(lp) : "memory");
        // S_ENDPGM performs an implicit wait-idle, covering the async store.
    }
}

extern "C" void kernel_launch(void* const* d_in, const int* in_sizes, int n_in,
                              void* d_out, int out_size, void* d_ws, size_t ws_size,
                              hipStream_t stream) {
    (void)n_in; (void)out_size; (void)d_ws; (void)ws_size;
    const int*   x = (const int*)d_in[0];
    const float* W = (const float*)d_in[1];
    float*     out = (float*)d_out;
    const int n_tokens = in_sizes[0];            // 32*4096 = 131072
    const int grid = (n_tokens + 1) / 2;         // 2 tokens per 256-thread block
    hipLaunchKernelGGL(embed_kernel, dim3(grid), dim3(256), 0, stream,
                       x, W, out, n_tokens);
}